// SPDBatchNormImpl_73443940761621
// MI455X (gfx1250) — compile-verified
//
#include <hip/hip_runtime.h>
#include <math.h>

#define Bn 4096
#define Cn 8
#define CLIPV 1e-6f
#define EPSV  1e-5f
#define ETAV  1.0f
#define NSWEEP 8

typedef float v2f __attribute__((ext_vector_type(2)));
typedef float v8f __attribute__((ext_vector_type(8)));

// ---------------------------------------------------------------------------
// Wave-level 32x32 f32 GEMM using V_WMMA_F32_16X16X4_F32 (wave32, EXEC all 1s)
// C = A * B ; all matrices row-major stride 32 in LDS.
// A fragment (16x4): lane l: m=l%16, VGPR j <-> K = 2*(l/16)+j
// B fragment (4x16): VGPR j holds row K=2*(l/16)+j, col = l%16
// C/D (16x16 f32): VGPR v of lane l holds row v+8*(l/16), col l%16
// ---------------------------------------------------------------------------
__device__ __forceinline__ void wave_mm32(const float* As, const float* Bs,
                                          float* Cs, int lane) {
  const int half = lane >> 4;
  const int l16  = lane & 15;
#pragma unroll
  for (int ti = 0; ti < 2; ++ti) {
#pragma unroll
    for (int tj = 0; tj < 2; ++tj) {
      v8f acc;
#pragma unroll
      for (int v = 0; v < 8; ++v) acc[v] = 0.0f;
#pragma unroll
      for (int k0 = 0; k0 < 32; k0 += 4) {
        const int ka = k0 + 2 * half;
        v2f a, b;
        a.x = As[(ti * 16 + l16) * 32 + ka + 0];
        a.y = As[(ti * 16 + l16) * 32 + ka + 1];
        b.x = Bs[(ka + 0) * 32 + tj * 16 + l16];
        b.y = Bs[(ka + 1) * 32 + tj * 16 + l16];
        acc = __builtin_amdgcn_wmma_f32_16x16x4_f32(false, a, false, b,
                                                    (short)0, acc, false, false);
      }
#pragma unroll
      for (int v = 0; v < 8; ++v)
        Cs[(ti * 16 + v + 8 * half) * 32 + tj * 16 + l16] = acc[v];
    }
  }
  __syncthreads();
}

// C = A * B^T (same layouts; B read transposed)
__device__ __forceinline__ void wave_mm32_ABt(const float* As, const float* Bs,
                                              float* Cs, int lane) {
  const int half = lane >> 4;
  const int l16  = lane & 15;
#pragma unroll
  for (int ti = 0; ti < 2; ++ti) {
#pragma unroll
    for (int tj = 0; tj < 2; ++tj) {
      v8f acc;
#pragma unroll
      for (int v = 0; v < 8; ++v) acc[v] = 0.0f;
#pragma unroll
      for (int k0 = 0; k0 < 32; k0 += 4) {
        const int ka = k0 + 2 * half;
        v2f a, b;
        a.x = As[(ti * 16 + l16) * 32 + ka + 0];
        a.y = As[(ti * 16 + l16) * 32 + ka + 1];
        b.x = Bs[(tj * 16 + l16) * 32 + ka + 0];   // B^T[ka][col] = B[col][ka]
        b.y = Bs[(tj * 16 + l16) * 32 + ka + 1];
        acc = __builtin_amdgcn_wmma_f32_16x16x4_f32(false, a, false, b,
                                                    (short)0, acc, false, false);
      }
#pragma unroll
      for (int v = 0; v < 8; ++v)
        Cs[(ti * 16 + v + 8 * half) * 32 + tj * 16 + l16] = acc[v];
    }
  }
  __syncthreads();
}

// ---------------------------------------------------------------------------
// One-sided cyclic Jacobi eigensolver, one wave per 32x32 symmetric matrix.
// On exit: G = A * V (columns = lam_j * v_j), V orthogonal eigenvectors.
// red: 96+ floats of LDS scratch per wave.
// ---------------------------------------------------------------------------
__device__ __forceinline__ void wave_jacobi(float* G, float* V, float* red,
                                            int lane) {
  for (int r = 0; r < 32; ++r) V[r * 32 + lane] = (r == lane) ? 1.f : 0.f;
  __syncthreads();
  const int half = lane >> 4;
  const int k    = lane & 15;
  for (int sweep = 0; sweep < NSWEEP; ++sweep) {
    for (int round = 0; round < 31; ++round) {
      // round-robin tournament pairing (16 disjoint pairs per round)
      int pp = (k == 0) ? 0 : (1 + (k - 1 + round) % 31);
      int qq = 1 + (30 - k + round) % 31;
      int p = pp < qq ? pp : qq;
      int q = pp < qq ? qq : pp;
      // column dot products; lanes k and k+16 each handle 16 rows
      float app = 0.f, aqq = 0.f, apq = 0.f;
      const int r0 = half * 16;
      for (int r = r0; r < r0 + 16; ++r) {
        float gp = G[r * 32 + p], gq = G[r * 32 + q];
        app = fmaf(gp, gp, app);
        aqq = fmaf(gq, gq, aqq);
        apq = fmaf(gp, gq, apq);
      }
      red[lane] = app; red[32 + lane] = aqq; red[64 + lane] = apq;
      __syncthreads();
      float APP = red[k] + red[k + 16];
      float AQQ = red[32 + k] + red[32 + k + 16];
      float APQ = red[64 + k] + red[64 + k + 16];
      float c = 1.f, s = 0.f;
      float thr = 1e-12f * sqrtf(APP * AQQ) + 1e-30f;
      if (fabsf(APQ) > thr) {
        float tau = (AQQ - APP) / (2.f * APQ);
        float t = ((tau >= 0.f) ? 1.f : -1.f) /
                  (fabsf(tau) + sqrtf(1.f + tau * tau));
        c = 1.f / sqrtf(1.f + t * t);
        s = t * c;
      }
      // apply Givens rotation to columns p,q of G and V (rows split by half)
      for (int r = r0; r < r0 + 16; ++r) {
        float gp = G[r * 32 + p], gq = G[r * 32 + q];
        G[r * 32 + p] = fmaf(c, gp, -s * gq);
        G[r * 32 + q] = fmaf(s, gp,  c * gq);
        float vp = V[r * 32 + p], vq = V[r * 32 + q];
        V[r * 32 + p] = fmaf(c, vp, -s * vq);
        V[r * 32 + q] = fmaf(s, vp,  c * vq);
      }
      __syncthreads();
    }
  }
}

// signed eigenvalue for column `lane`: lam_j = g_j . v_j
__device__ __forceinline__ float wave_coldot(const float* G, const float* V,
                                             int lane) {
  float d = 0.f;
  for (int r = 0; r < 32; ++r)
    d = fmaf(G[r * 32 + lane], V[r * 32 + lane], d);
  return d;
}

// OUT = V * diag(f_lane) * V^T   (W = scratch)
__device__ __forceinline__ void wave_recon(const float* V, float* W, float* OUT,
                                           float f, int lane) {
  for (int r = 0; r < 32; ++r) W[r * 32 + lane] = V[r * 32 + lane] * f;
  __syncthreads();
  wave_mm32_ABt(W, V, OUT, lane);
}

__device__ __forceinline__ void wave_load1024(const float* __restrict__ g,
                                              float* s, int lane) {
#pragma unroll
  for (int i = 0; i < 8; ++i)
    *(float4*)(s + i * 128 + lane * 4) =
        *(const float4*)(g + i * 128 + lane * 4);
  __syncthreads();
}

__device__ __forceinline__ void wave_store1024(float* __restrict__ g,
                                               const float* s, int lane) {
#pragma unroll
  for (int i = 0; i < 8; ++i)
    *(float4*)(g + i * 128 + lane * 4) =
        *(const float4*)(s + i * 128 + lane * 4);
}

// ---------------------------------------------------------------------------
__global__ void k_zero(float* p, int n) {
  int i = blockIdx.x * 256 + threadIdx.x;
  if (i < n) p[i] = 0.f;
}

// partial batch-mean: grid (32 chunks, C), block 256
__global__ void k_redu1(const float* __restrict__ X, float* __restrict__ partial) {
  int chunk = blockIdx.x, c = blockIdx.y, t = threadIdx.x;
  int b0 = chunk * (Bn / 32);
  for (int e = t; e < 1024; e += 256) {
    float s = 0.f;
    for (int i = 0; i < Bn / 32; ++i)
      s += X[((size_t)((b0 + i) * Cn + c)) * 1024 + e];
    partial[(c * 32 + chunk) * 1024 + e] = s;
  }
}

__global__ void k_redu2(const float* __restrict__ partial, float* __restrict__ bm) {
  int c = blockIdx.x, t = threadIdx.x;
  for (int e = t; e < 1024; e += 256) {
    float s = 0.f;
    for (int k = 0; k < 32; ++k) s += partial[(c * 32 + k) * 1024 + e];
    bm[c * 1024 + e] = s * (1.f / Bn);
  }
}

// per-channel eigh of batch mean -> bm^{1/2}, bm^{-1/2}
__global__ __launch_bounds__(32) void k_bm_eig(const float* __restrict__ bm,
                                               float* __restrict__ bm_sq,
                                               float* __restrict__ bm_isq) {
  __shared__ float sm[4 * 1024 + 128];
  float* G = sm;            float* V = sm + 1024;
  float* W = sm + 2048;     float* O = sm + 3072;
  float* red = sm + 4096;
  int lane = threadIdx.x, c = blockIdx.x;
  wave_load1024(bm + c * 1024, G, lane);
  wave_jacobi(G, V, red, lane);
  float lam = fmaxf(wave_coldot(G, V, lane), CLIPV);
  wave_recon(V, W, O, sqrtf(lam), lane);
  wave_store1024(bm_sq + c * 1024, O, lane);
  __syncthreads();
  wave_recon(V, W, O, 1.f / sqrtf(lam), lane);
  wave_store1024(bm_isq + c * 1024, O, lane);
}

// phase 1: XT = logm(isq X isq); accumulate GT sum and ||XT||^2 sum
__global__ __launch_bounds__(64) void k_phase1(const float* __restrict__ X,
                                               const float* __restrict__ bm_isq,
                                               float* __restrict__ GTsum,
                                               float* __restrict__ froSum) {
  __shared__ float sm[2 * 3200 + 1024];
  __shared__ float accF;
  int wave = threadIdx.x >> 5, lane = threadIdx.x & 31;
  float* M   = sm + wave * 3200;
  float* Xb  = M + 1024;
  float* T   = M + 2048;
  float* red = M + 3072;           // 128
  float* acc = sm + 2 * 3200;      // 1024 block accumulator
  int c = blockIdx.y;
  int b = blockIdx.x * 2 + wave;

  for (int i = threadIdx.x; i < 1024; i += 64) acc[i] = 0.f;
  if (threadIdx.x == 0) accF = 0.f;
  __syncthreads();

  wave_load1024(bm_isq + c * 1024, M, lane);
  wave_load1024(X + ((size_t)(b * Cn + c)) * 1024, Xb, lane);
  wave_mm32(M, Xb, T, lane);          // T = isq * X
  wave_mm32(T, M, Xb, lane);          // Xb = Y = isq X isq
  wave_jacobi(Xb, T, red, lane);      // G=Xb, V=T
  float lam = wave_coldot(Xb, T, lane);
  float f = logf(fmaxf(lam, CLIPV));
  wave_recon(T, M, Xb, f, lane);      // Xb = XT (M scratch)

  float fr = 0.f;
  for (int r = 0; r < 32; ++r) {
    float x = Xb[r * 32 + lane];
    fr = fmaf(x, x, fr);
    atomicAdd(&acc[r * 32 + lane], x);     // LDS atomic
  }
  red[lane] = fr;
  __syncthreads();
  if (lane == 0) {
    float s = 0.f;
    for (int i = 0; i < 32; ++i) s += red[i];
    atomicAdd(&accF, s);
  }
  __syncthreads();
  for (int i = threadIdx.x; i < 1024; i += 64)
    atomicAdd(&GTsum[c * 1024 + i], acc[i]);
  if (threadIdx.x == 0) atomicAdd(&froSum[c], accF);
}

// per-channel: expm/geodesic/GT2/variance/scale + rm^{-1/2}, mean^{1/2}
__global__ __launch_bounds__(32) void k_channel(
    const float* __restrict__ meanG, const float* __restrict__ stdv,
    const float* __restrict__ run_mean, const float* __restrict__ run_var,
    const float* __restrict__ bm_sq, const float* __restrict__ bm_isq,
    const float* __restrict__ GTsum, const float* __restrict__ froSum,
    float* __restrict__ rm_isq_o, float* __restrict__ m_sq_o,
    float* __restrict__ sval_o) {
  __shared__ float sm[11 * 1024 + 128];
  float* GTm = sm;              float* S1  = sm + 1024;
  float* S2  = sm + 2048;       float* G   = sm + 3072;
  float* V   = sm + 4096;       float* W   = sm + 5120;
  float* T   = sm + 6144;       float* T2  = sm + 7168;
  float* RM  = sm + 8192;       float* RSQ = sm + 9216;
  float* OUT = sm + 10240;      float* red = sm + 11264;
  int lane = threadIdx.x, c = blockIdx.x;

  for (int r = 0; r < 32; ++r)
    GTm[r * 32 + lane] = GTsum[c * 1024 + r * 32 + lane] * (1.f / Bn);
  wave_load1024(bm_sq + c * 1024, S1, lane);
  wave_load1024(bm_isq + c * 1024, S2, lane);

  // bm_new = S1 * expm(GT) * S1
  for (int r = 0; r < 32; ++r) G[r * 32 + lane] = GTm[r * 32 + lane];
  __syncthreads();
  wave_jacobi(G, V, red, lane);
  float lamE = wave_coldot(G, V, lane);     // signed! GT is indefinite
  wave_recon(V, W, T, expf(lamE), lane);    // T = expm(GT)
  wave_mm32(S1, T, T2, lane);
  wave_mm32(T2, S1, T, lane);               // T = bm_new

  // geodesic: rm = rsq * expm(eta*logm(risq bm_new risq)) * rsq
  wave_load1024(run_mean + c * 1024, G, lane);
  wave_jacobi(G, V, red, lane);
  float lamR = fmaxf(wave_coldot(G, V, lane), CLIPV);
  wave_recon(V, W, RSQ, sqrtf(lamR), lane);       // RSQ = rm^{1/2}
  wave_recon(V, W, G, 1.f / sqrtf(lamR), lane);   // G   = rm^{-1/2}
  wave_mm32(G, T, T2, lane);
  wave_mm32(T2, G, T, lane);                // T = Z
  wave_jacobi(T, V, red, lane);
  float lamZ = wave_coldot(T, V, lane);
  float fZ = expf(ETAV * logf(fmaxf(lamZ, CLIPV)));
  wave_recon(V, W, T2, fZ, lane);
  wave_mm32(RSQ, T2, T, lane);
  wave_mm32(T, RSQ, RM, lane);              // RM = rm

  // GT2 = logm(S2 rm S2)
  wave_mm32(S2, RM, T, lane);
  wave_mm32(T, S2, T2, lane);
  wave_jacobi(T2, V, red, lane);
  float lamG = wave_coldot(T2, V, lane);
  wave_recon(V, W, OUT, logf(fmaxf(lamG, CLIPV)), lane);   // OUT = GT2

  // batch_var = mean||XT||^2 - 2<GT,GT2> + ||GT2||^2 ; s = std/sqrt(rv+eps)
  float dotp = 0.f, n2 = 0.f;
  for (int r = 0; r < 32; ++r) {
    float g2 = OUT[r * 32 + lane];
    dotp = fmaf(GTm[r * 32 + lane], g2, dotp);
    n2 = fmaf(g2, g2, n2);
  }
  red[lane] = dotp; red[32 + lane] = n2;
  __syncthreads();
  if (lane == 0) {
    float d = 0.f, nn = 0.f;
    for (int i = 0; i < 32; ++i) { d += red[i]; nn += red[32 + i]; }
    float froXT = froSum[c] * (1.f / Bn);
    float bv = froXT - 2.f * d + nn;
    float rv = (1.f - ETAV) * run_var[c] + ETAV * bv;
    sval_o[c] = stdv[c] / sqrtf(rv + EPSV);
  }
  __syncthreads();

  // rm^{-1/2}
  wave_jacobi(RM, V, red, lane);
  float lamM = fmaxf(wave_coldot(RM, V, lane), CLIPV);
  wave_recon(V, W, OUT, 1.f / sqrtf(lamM), lane);
  wave_store1024(rm_isq_o + c * 1024, OUT, lane);
  __syncthreads();

  // mean^{1/2}
  wave_load1024(meanG + c * 1024, G, lane);
  wave_jacobi(G, V, red, lane);
  float lamQ = fmaxf(wave_coldot(G, V, lane), CLIPV);
  wave_recon(V, W, OUT, sqrtf(lamQ), lane);
  wave_store1024(m_sq_o + c * 1024, OUT, lane);
}

// phase 2: out = m_sq * (rm_isq X rm_isq)^s * m_sq
__global__ __launch_bounds__(64) void k_phase2(const float* __restrict__ X,
                                               const float* __restrict__ rm_isq,
                                               const float* __restrict__ m_sq,
                                               const float* __restrict__ sval,
                                               float* __restrict__ out) {
  __shared__ float sm[2 * 4224];
  int wave = threadIdx.x >> 5, lane = threadIdx.x & 31;
  float* M   = sm + wave * 4224;
  float* M2  = M + 1024;
  float* Xb  = M + 2048;
  float* T   = M + 3072;
  float* red = M + 4096;          // 128
  int c = blockIdx.y;
  int b = blockIdx.x * 2 + wave;
  float s = sval[c];
  wave_load1024(rm_isq + c * 1024, M, lane);
  wave_load1024(m_sq + c * 1024, M2, lane);
  wave_load1024(X + ((size_t)(b * Cn + c)) * 1024, Xb, lane);
  wave_mm32(M, Xb, T, lane);
  wave_mm32(T, M, Xb, lane);          // Xb = Z
  wave_jacobi(Xb, T, red, lane);      // G=Xb, V=T
  float lam = wave_coldot(Xb, T, lane);
  float f = expf(s * logf(fmaxf(lam, CLIPV)));   // clip(lam)^s
  wave_recon(T, M, Xb, f, lane);      // Xb = P
  wave_mm32(M2, Xb, T, lane);         // T = m_sq * P
  wave_mm32(T, M2, Xb, lane);         // Xb = result
  wave_store1024(out + ((size_t)(b * Cn + c)) * 1024, Xb, lane);
}

// ---------------------------------------------------------------------------
extern "C" void kernel_launch(void* const* d_in, const int* in_sizes, int n_in,
                              void* d_out, int out_size, void* d_ws,
                              size_t ws_size, hipStream_t stream) {
  (void)in_sizes; (void)n_in; (void)out_size; (void)ws_size;
  const float* X     = (const float*)d_in[0];
  const float* meanG = (const float*)d_in[1];
  const float* stdv  = (const float*)d_in[2];
  const float* rmean = (const float*)d_in[3];
  const float* rvar  = (const float*)d_in[4];
  float* out = (float*)d_out;
  float* ws  = (float*)d_ws;

  float* partial = ws;                       // 8*32*1024
  float* bm      = partial + Cn * 32 * 1024; // 8192
  float* bm_sq   = bm + Cn * 1024;
  float* bm_isq  = bm_sq + Cn * 1024;
  float* GTsum   = bm_isq + Cn * 1024;       // 8192 (+32 froSum right after)
  float* froSum  = GTsum + Cn * 1024;        // 32
  float* rm_isq  = froSum + 32;
  float* m_sq    = rm_isq + Cn * 1024;
  float* sval    = m_sq + Cn * 1024;

  k_zero<<<dim3((Cn * 1024 + 32 + 255) / 256), 256, 0, stream>>>(GTsum,
                                                                 Cn * 1024 + 32);
  k_redu1<<<dim3(32, Cn), 256, 0, stream>>>(X, partial);
  k_redu2<<<dim3(Cn), 256, 0, stream>>>(partial, bm);
  k_bm_eig<<<dim3(Cn), 32, 0, stream>>>(bm, bm_sq, bm_isq);
  k_phase1<<<dim3(Bn / 2, Cn), 64, 0, stream>>>(X, bm_isq, GTsum, froSum);
  k_channel<<<dim3(Cn), 32, 0, stream>>>(meanG, stdv, rmean, rvar, bm_sq,
                                         bm_isq, GTsum, froSum, rm_isq, m_sq,
                                         sval);
  k_phase2<<<dim3(Bn / 2, Cn), 64, 0, stream>>>(X, rm_isq, m_sq, sval, out);
}